// WaveConv3d_40072044872091
// MI455X (gfx1250) — compile-verified
//
#include <hip/hip_runtime.h>

// MI455X / gfx1250, wave32. Memory-bound pipeline (~1 GiB HBM traffic => ~44us
// floor @ 23.3 TB/s); spectral compute (~5 GFLOP) done exactly in fp32 with
// V_WMMA_F32_16X16X4_F32 for the per-mode complex einsum.

typedef __attribute__((ext_vector_type(2))) float v2f;
typedef __attribute__((ext_vector_type(8))) float v8f;

#define INV_SQRT8 0.35355339059327373f   // (1/sqrt(2))^3

__constant__ float COS16[16] = {
    1.0f,  0.9238795325112867f,  0.7071067811865476f,  0.3826834323650898f,
    0.0f, -0.3826834323650898f, -0.7071067811865476f, -0.9238795325112867f,
   -1.0f, -0.9238795325112867f, -0.7071067811865476f, -0.3826834323650898f,
    0.0f,  0.3826834323650898f,  0.7071067811865476f,  0.9238795325112867f};
__constant__ float SIN16[16] = {
    0.0f,  0.3826834323650898f,  0.7071067811865476f,  0.9238795325112867f,
    1.0f,  0.9238795325112867f,  0.7071067811865476f,  0.3826834323650898f,
    0.0f, -0.3826834323650898f, -0.7071067811865476f, -0.9238795325112867f,
   -1.0f, -0.9238795325112867f, -0.7071067811865476f, -0.3826834323650898f};

// ---------------- workspace layout (floats) ----------------
// d1  : L1 details [7][4][64][32][32][32]            = 58,720,256
// l2  : L2 coeffs  [8][4][64][16][16][16]            =  8,388,608  (reused as spec output)
// xh  : A-frags    [8][4cn][27m][2ri][64 i][16 bpad] =  1,769,472
// oh  : einsum out [8][4cn][27m][2ri][4 b][64 o]     =    442,368
#define D1_SB   8388608LL          // per-subband stride of d1 (4*64*32^3)
#define L1_SZ   58720256LL
#define L2_SZ   8388608LL
#define XH_SZ   1769472LL
#define OH_SZ   442368LL

__global__ __launch_bounds__(256) void k_zero(float* __restrict__ p, long long n) {
  long long i = (long long)blockIdx.x * blockDim.x + threadIdx.x;
  if (i < n) p[i] = 0.0f;
}

// ---------------- K1: fused 2-level 3D Haar DWT ----------------
// grid: 8192 = b(4) * c(64) * xb(16) * yb(2); block 256
__global__ __launch_bounds__(256) void k_dwt2(const float* __restrict__ x,
                                              float* __restrict__ d1,
                                              float* __restrict__ l2) {
  __shared__ float tile[4 * 32 * 64];  // [dx4][y32][z64]  32 KB
  __shared__ float a1[2 * 16 * 32];    // L1 aaa  [X'2][y'16][z'32]  4 KB
  int t = threadIdx.x;
  int bid = blockIdx.x;
  int yb = bid & 1;
  int xb = (bid >> 1) & 15;
  int c  = (bid >> 5) & 63;
  int b  = bid >> 11;

  const float* xp = x + (((long long)(b * 64 + c) * 64 + xb * 4) * 64 + yb * 32) * 64;
  for (int i = t; i < 8192; i += 256) {
    int dx = i >> 11; int rem = i & 2047;
    tile[i] = xp[dx * 4096 + rem];
  }
  __syncthreads();

  // level 1: coarse positions (X'loc 2) x (y'loc 16) x (z' 32) = 1024
  for (int p = t; p < 1024; p += 256) {
    int dxp = p >> 9; int yl = (p >> 5) & 15; int zp = p & 31;
    const float* q = tile + (2 * dxp) * 2048 + (2 * yl) * 64 + 2 * zp;
    float v000 = q[0],    v001 = q[1],    v010 = q[64],   v011 = q[65];
    float v100 = q[2048], v101 = q[2049], v110 = q[2112], v111 = q[2113];
    float az00 = v000 + v001, dz00 = v000 - v001;
    float az01 = v010 + v011, dz01 = v010 - v011;
    float az10 = v100 + v101, dz10 = v100 - v101;
    float az11 = v110 + v111, dz11 = v110 - v111;
    float aa0 = az00 + az01, da0 = az00 - az01, ad0 = dz00 + dz01, dd0 = dz00 - dz01;
    float aa1 = az10 + az11, da1 = az10 - az11, ad1 = dz10 + dz11, dd1 = dz10 - dz11;
    float o0 = aa0 + aa1, o4 = aa0 - aa1;
    float o2 = da0 + da1, o6 = da0 - da1;
    float o1 = ad0 + ad1, o5 = ad0 - ad1;
    float o3 = dd0 + dd1, o7 = dd0 - dd1;
    long long dbase = ((long long)(b * 64 + c)) * 32768 +
                      (long long)(xb * 2 + dxp) * 1024 + (yb * 16 + yl) * 32 + zp;
    d1[0 * D1_SB + dbase] = o1 * INV_SQRT8;  // aad
    d1[1 * D1_SB + dbase] = o2 * INV_SQRT8;  // ada
    d1[2 * D1_SB + dbase] = o3 * INV_SQRT8;  // add
    d1[3 * D1_SB + dbase] = o4 * INV_SQRT8;  // daa
    d1[4 * D1_SB + dbase] = o5 * INV_SQRT8;  // dad
    d1[5 * D1_SB + dbase] = o6 * INV_SQRT8;  // dda
    d1[6 * D1_SB + dbase] = o7 * INV_SQRT8;  // ddd
    a1[p] = o0 * INV_SQRT8;                  // aaa -> level 2
  }
  __syncthreads();

  // level 2 on a1: (y''loc 8) x (z'' 16) = 128
  if (t < 128) {
    int yl2 = t >> 4; int z2 = t & 15;
    const float* q = a1 + (2 * yl2) * 32 + 2 * z2;
    float v000 = q[0],   v001 = q[1],   v010 = q[32],  v011 = q[33];
    float v100 = q[512], v101 = q[513], v110 = q[544], v111 = q[545];
    float az00 = v000 + v001, dz00 = v000 - v001;
    float az01 = v010 + v011, dz01 = v010 - v011;
    float az10 = v100 + v101, dz10 = v100 - v101;
    float az11 = v110 + v111, dz11 = v110 - v111;
    float aa0 = az00 + az01, da0 = az00 - az01, ad0 = dz00 + dz01, dd0 = dz00 - dz01;
    float aa1 = az10 + az11, da1 = az10 - az11, ad1 = dz10 + dz11, dd1 = dz10 - dz11;
    float o[8];
    o[0] = aa0 + aa1; o[4] = aa0 - aa1;
    o[2] = da0 + da1; o[6] = da0 - da1;
    o[1] = ad0 + ad1; o[5] = ad0 - ad1;
    o[3] = dd0 + dd1; o[7] = dd0 - dd1;
    long long sp = (long long)xb * 256 + (yb * 8 + yl2) * 16 + z2;
    for (int s = 0; s < 8; ++s)
      l2[(((long long)s * 4 + b) * 64 + c) * 4096 + sp] = o[s] * INV_SQRT8;
  }
}

// ---------------- K2: forward mode projection (separable DFT to 108 modes) ----
// grid: 2048 = s(8)*b(4)*c(64); block 256
__global__ __launch_bounds__(256) void k_fwd_modes(const float* __restrict__ l2,
                                                   float* __restrict__ xh) {
  __shared__ float tile[4096];
  __shared__ float T1r[768], T1i[768];   // [x16][y16][kz3]
  __shared__ float T2r[288], T2i[288];   // [x16][ky6][kz3]
  int t = threadIdx.x;
  int bid = blockIdx.x;
  int s = bid >> 8; int b = (bid >> 6) & 3; int c = bid & 63;
  const float* src = l2 + (((long long)s * 4 + b) * 64 + c) * 4096;
  for (int i = t; i < 4096; i += 256) tile[i] = src[i];
  __syncthreads();

  {  // z transform: 256 threads, one (x,y) each
    int x = t >> 4, y = t & 15;
    const float* row = tile + x * 256 + y * 16;
    for (int kz = 0; kz < 3; ++kz) {
      float cr = 0.f, ci = 0.f;
      for (int z = 0; z < 16; ++z) {
        float v = row[z]; int ph = (kz * z) & 15;
        cr += v * COS16[ph]; ci -= v * SIN16[ph];
      }
      T1r[t * 3 + kz] = cr; T1i[t * 3 + kz] = ci;
    }
  }
  __syncthreads();

  for (int idx = t; idx < 288; idx += 256) {  // y transform
    int x = idx / 18; int r = idx % 18; int kyi = r / 3; int kz = r % 3;
    int ky = (kyi < 3) ? kyi : (10 + kyi);  // 13,14,15
    float cr = 0.f, ci = 0.f;
    for (int y = 0; y < 16; ++y) {
      float ar = T1r[(x * 16 + y) * 3 + kz], ai = T1i[(x * 16 + y) * 3 + kz];
      int ph = (ky * y) & 15; float cc = COS16[ph], ss = SIN16[ph];
      cr += ar * cc + ai * ss;   // (ar+i*ai)*(cc - i*ss)
      ci += ai * cc - ar * ss;
    }
    T2r[idx] = cr; T2i[idx] = ci;
  }
  __syncthreads();

  if (t < 108) {  // x transform + scatter into WMMA A-fragment layout
    int kxi = t / 18; int r = t % 18; int kyi = r / 3; int kz = r % 3;
    int kx = (kxi < 3) ? kxi : (10 + kxi);
    float cr = 0.f, ci = 0.f;
    for (int x = 0; x < 16; ++x) {
      float ar = T2r[x * 18 + kyi * 3 + kz], ai = T2i[x * 18 + kyi * 3 + kz];
      int ph = (kx * x) & 15; float cc = COS16[ph], ss = SIN16[ph];
      cr += ar * cc + ai * ss;
      ci += ai * cc - ar * ss;
    }
    int cn = ((kxi >= 3) ? 1 : 0) | ((kyi >= 3) ? 2 : 0);
    int m = ((kxi % 3) * 3 + (kyi % 3)) * 3 + kz;
    long long base = ((((long long)s * 4 + cn) * 27 + m) * 2) * 1024 + c * 16 + b;
    xh[base]        = cr;   // re
    xh[base + 1024] = ci;   // im
  }
}

// ---------------- K3: per-mode complex einsum via V_WMMA_F32_16X16X4_F32 ------
// grid: 864 = s(8)*cn(4)*m(27); block 128 (4 waves, one 16-out-channel tile each)
__global__ __launch_bounds__(128) void k_einsum_wmma(const float* __restrict__ xh,
                                                     const float* __restrict__ w,
                                                     float* __restrict__ oh) {
  int bid = blockIdx.x;
  int s = bid / 108; int r = bid % 108; int cn = r / 27; int m = r % 27;
  int mx = m / 9, my = (m / 3) % 3, mz = m % 3;
  int tid = threadIdx.x; int lane = tid & 31; int wv = tid >> 5;
  int n0 = wv * 16;
  int row = lane & 15;
  int khalf = (lane < 16) ? 0 : 2;

  const float* XA = xh + ((((long long)s * 4 + cn) * 27 + m) * 2) * 1024;
  // weights[8][4][64 i][64 o][3][3][3][2]
  const float* WB = w + (long long)s * 884736 + (long long)cn * 221184 +
                    (long long)(n0 + row) * 54 + mx * 18 + my * 6 + mz * 2;

  v8f cr = {0.f, 0.f, 0.f, 0.f, 0.f, 0.f, 0.f, 0.f};
  v8f ci = {0.f, 0.f, 0.f, 0.f, 0.f, 0.f, 0.f, 0.f};
  for (int k0 = 0; k0 < 64; k0 += 4) {
    int iA = k0 + khalf;
    v2f ar, ai, br, bi;
    ar.x = XA[iA * 16 + row];          ar.y = XA[(iA + 1) * 16 + row];
    ai.x = XA[1024 + iA * 16 + row];   ai.y = XA[1024 + (iA + 1) * 16 + row];
    v2f nai = -ai;
    br.x = WB[(long long)iA * 3456];        br.y = WB[(long long)(iA + 1) * 3456];
    bi.x = WB[(long long)iA * 3456 + 1];    bi.y = WB[(long long)(iA + 1) * 3456 + 1];
    // Cr += xr*wr - xi*wi ; Ci += xr*wi + xi*wr
    cr = __builtin_amdgcn_wmma_f32_16x16x4_f32(false, ar,  false, br, (short)0, cr, false, false);
    cr = __builtin_amdgcn_wmma_f32_16x16x4_f32(false, nai, false, bi, (short)0, cr, false, false);
    ci = __builtin_amdgcn_wmma_f32_16x16x4_f32(false, ar,  false, bi, (short)0, ci, false, false);
    ci = __builtin_amdgcn_wmma_f32_16x16x4_f32(false, ai,  false, br, (short)0, ci, false, false);
  }
  if (lane < 16) {  // rows M=0..3 live in C VGPRs 0..3, lanes 0..15
    long long ob = ((((long long)s * 4 + cn) * 27 + m) * 2) * 256;
    int o = n0 + lane;
    for (int j = 0; j < 4; ++j) {
      oh[ob + j * 64 + o]       = cr[j];
      oh[ob + 256 + j * 64 + o] = ci[j];
    }
  }
}

// ---------------- K4: inverse mode synthesis (irfftn on 108 modes) ------------
// grid: 2048 = s(8)*b(4)*o(64); block 256
__global__ __launch_bounds__(256) void k_inv_modes(const float* __restrict__ oh,
                                                   float* __restrict__ spec) {
  __shared__ float Om[216];  // [cn4][m27][ri2]
  int t = threadIdx.x;
  int bid = blockIdx.x;
  int s = bid >> 8; int b = (bid >> 6) & 3; int o = bid & 63;
  if (t < 216) {
    int cn = t / 54; int r = t % 54; int m = r >> 1; int ri = r & 1;
    Om[t] = oh[(((((long long)s * 4 + cn) * 27 + m) * 2 + ri) * 4 + b) * 64 + o];
  }
  __syncthreads();

  int x = t >> 4, y = t & 15;
  float Pr0 = 0.f, Pi0 = 0.f, Pr1 = 0.f, Pi1 = 0.f, Pr2 = 0.f, Pi2 = 0.f;
  for (int cn = 0; cn < 4; ++cn)
    for (int mx = 0; mx < 3; ++mx) {
      int kx = (cn & 1) ? (13 + mx) : mx;
      for (int my = 0; my < 3; ++my) {
        int ky = (cn & 2) ? (13 + my) : my;
        int ph = (kx * x + ky * y) & 15;
        float cc = COS16[ph], ss = SIN16[ph];
        int mb = (cn * 27 + (mx * 3 + my) * 3) * 2;
        float Or, Oi;
        Or = Om[mb + 0]; Oi = Om[mb + 1];
        Pr0 += Or * cc - Oi * ss;  Pi0 += Or * ss + Oi * cc;
        Or = Om[mb + 2]; Oi = Om[mb + 3];
        Pr1 += Or * cc - Oi * ss;  Pi1 += Or * ss + Oi * cc;
        Or = Om[mb + 4]; Oi = Om[mb + 5];
        Pr2 += Or * cc - Oi * ss;  Pi2 += Or * ss + Oi * cc;
      }
    }
  float* dst = spec + (((long long)s * 4 + b) * 64 + o) * 4096 + x * 256 + y * 16;
  const float inv = 1.0f / 4096.0f;
  for (int z = 0; z < 16; ++z) {
    float v = Pr0;  // kz=0: c2r uses Re only, weight 1
    v += 2.f * (Pr1 * COS16[z] - Pi1 * SIN16[z]);                 // kz=1, Hermitian x2
    v += 2.f * (Pr2 * COS16[(2 * z) & 15] - Pi2 * SIN16[(2 * z) & 15]);  // kz=2
    dst[z] = v * inv;
  }
}

// ---------------- K5: fused 2-level inverse Haar DWT --------------------------
// grid: 4096 = b(4)*co(64)*xb(16); block 256
__global__ __launch_bounds__(256) void k_idwt2(const float* __restrict__ spec,
                                               const float* __restrict__ d1,
                                               float* __restrict__ out) {
  __shared__ float ls[8 * 256];  // [s][y''16][z''16]
  __shared__ float a1[2048];     // [X'2][y'32][z'32]
  int t = threadIdx.x;
  int bid = blockIdx.x;
  int xb = bid & 15; int co = (bid >> 4) & 63; int b = bid >> 10;

  for (int i = t; i < 2048; i += 256) {
    int s = i >> 8; int yz = i & 255;
    ls[i] = spec[(((long long)s * 4 + b) * 64 + co) * 4096 + (long long)xb * 256 + yz];
  }
  __syncthreads();

  // inverse level 2 -> L1 aaa
  for (int p = t; p < 2048; p += 256) {
    int dxp = p >> 10; int yp = (p >> 5) & 31; int zp = p & 31;
    int dbits = (dxp << 2) | ((yp & 1) << 1) | (zp & 1);
    const float* cb = ls + (yp >> 1) * 16 + (zp >> 1);
    float c0 = cb[0],    c1 = cb[256],  c2 = cb[512],  c3 = cb[768];
    float c4 = cb[1024], c5 = cb[1280], c6 = cb[1536], c7 = cb[1792];
    float e01 = c0 + ((dbits & 1) ? -c1 : c1);
    float e23 = c2 + ((dbits & 1) ? -c3 : c3);
    float e45 = c4 + ((dbits & 1) ? -c5 : c5);
    float e67 = c6 + ((dbits & 1) ? -c7 : c7);
    float f03 = e01 + ((dbits & 2) ? -e23 : e23);
    float f47 = e45 + ((dbits & 2) ? -e67 : e67);
    a1[p] = (f03 + ((dbits & 4) ? -f47 : f47)) * INV_SQRT8;
  }
  __syncthreads();

  // inverse level 1: coarse pos (X'loc 2, y'32, z'32); emit 2x2x2 outputs each
  for (int p = t; p < 2048; p += 256) {
    int dxp = p >> 10; int yp = (p >> 5) & 31; int zp = p & 31;
    long long dbase = (((long long)(b * 64 + co)) * 32 + (xb * 2 + dxp)) * 1024 + yp * 32 + zp;
    float c0 = a1[p];
    float c1 = d1[0 * D1_SB + dbase];
    float c2 = d1[1 * D1_SB + dbase];
    float c3 = d1[2 * D1_SB + dbase];
    float c4 = d1[3 * D1_SB + dbase];
    float c5 = d1[4 * D1_SB + dbase];
    float c6 = d1[5 * D1_SB + dbase];
    float c7 = d1[6 * D1_SB + dbase];
    float t0 = c0 + c4, t1 = c1 + c5, t2 = c2 + c6, t3 = c3 + c7;
    float u0 = c0 - c4, u1 = c1 - c5, u2 = c2 - c6, u3 = c3 - c7;
    float p0 = t0 + t2, p1 = t1 + t3, q0 = t0 - t2, q1 = t1 - t3;
    float r0 = u0 + u2, r1 = u1 + u3, w0 = u0 - u2, w1 = u1 - u3;
    float* ob = out + (((long long)(b * 64 + co)) * 64 + (xb * 4 + 2 * dxp)) * 4096 +
                (2 * yp) * 64 + 2 * zp;
    ob[0]    = (p0 + p1) * INV_SQRT8;
    ob[1]    = (p0 - p1) * INV_SQRT8;
    ob[64]   = (q0 + q1) * INV_SQRT8;
    ob[65]   = (q0 - q1) * INV_SQRT8;
    ob[4096] = (r0 + r1) * INV_SQRT8;
    ob[4097] = (r0 - r1) * INV_SQRT8;
    ob[4160] = (w0 + w1) * INV_SQRT8;
    ob[4161] = (w0 - w1) * INV_SQRT8;
  }
}

extern "C" void kernel_launch(void* const* d_in, const int* in_sizes, int n_in,
                              void* d_out, int out_size, void* d_ws, size_t ws_size,
                              hipStream_t stream) {
  (void)in_sizes; (void)n_in; (void)out_size; (void)ws_size;
  const float* x = (const float*)d_in[0];
  const float* w = (const float*)d_in[1];
  float* out = (float*)d_out;

  float* ws  = (float*)d_ws;
  float* d1  = ws;                 // 58,720,256 floats
  float* l2  = d1 + L1_SZ;         //  8,388,608 floats
  float* xh  = l2 + L2_SZ;         //  1,769,472 floats
  float* oh  = xh + XH_SZ;         //    442,368 floats
  float* spec = l2;                // alias: l2 fully consumed before spec written

  k_zero<<<(unsigned)((XH_SZ + 255) / 256), 256, 0, stream>>>(xh, XH_SZ);  // zero b-pad rows
  k_dwt2<<<8192, 256, 0, stream>>>(x, d1, l2);
  k_fwd_modes<<<2048, 256, 0, stream>>>(l2, xh);
  k_einsum_wmma<<<864, 128, 0, stream>>>(xh, w, oh);
  k_inv_modes<<<2048, 256, 0, stream>>>(oh, spec);
  k_idwt2<<<4096, 256, 0, stream>>>(spec, d1, out);
}